// NCCLoss_81638738362418
// MI455X (gfx1250) — compile-verified
//
#include <hip/hip_runtime.h>
#include <math.h>

typedef __attribute__((ext_vector_type(2))) float v2f;
typedef __attribute__((ext_vector_type(4))) float v4f;
typedef __attribute__((ext_vector_type(8))) float v8f;

#define NCC_W   160
#define NCC_H   192
#define NCC_D   160
#define NCC_B   2
#define NCC_HW  (NCC_H * NCC_W)          // 30720
#define NCC_DHW (NCC_D * NCC_H * NCC_W)  // 4915200
#define NCC_N   (NCC_B * NCC_DHW)        // 9830400 voxels
#define NCC_ROWS (NCC_B * NCC_D * NCC_H) // 61440 independent W-rows
#define WTILES  (NCC_W / 16)             // 10
#define HTILES  (NCC_H / 16)             // 12
#define ROWTILES (NCC_ROWS / 16)         // 3840
#define TILES_W (ROWTILES * WTILES)      // 38400 tiles, pass W
#define SLICES  (NCC_B * NCC_D)          // 320
#define TILES_H (SLICES * HTILES * WTILES) // 38400 tiles, pass H
#define COLS    (NCC_B * NCC_HW)         // 61440 columns, pass D
#define WIN_R   4
#define NCC_EPS 1e-5f

// D = A(16x4,f32) * B(4x16,f32) + C on the matrix pipe (wave32, EXEC all-1s).
__device__ __forceinline__ v8f wmma_f32_16x16x4(v2f a, v2f b, v8f c) {
  return __builtin_amdgcn_wmma_f32_16x16x4_f32(
      /*neg_a=*/false, a, /*neg_b=*/false, b,
      /*c_mod=*/(short)0, c, /*reuse_a=*/false, /*reuse_b=*/false);
}

// 1.0 iff |d| <= 4 (9-wide box window), single unsigned compare.
__device__ __forceinline__ float band9(int d) {
  return ((unsigned)(d + WIN_R) <= (unsigned)(2 * WIN_R)) ? 1.f : 0.f;
}

// ---------------------------------------------------------------------------
// Pass 1: per-voxel products + 9-wide box sum along W (stride-1 axis).
// One wave per 16(rows) x 16(w) tile; K spans [w0-4, w0+19] as 6 chunks of 4.
// Chunks 1..4 are always fully in range (16-aligned tiles, dims % 16 == 0);
// only the first/last chunk needs a whole-chunk validity flag, folded into
// the B band matrix, with a clamped-safe load address.
// Field outputs go to 5 separate kernarg base pointers -> saddr-form stores.
// ---------------------------------------------------------------------------
__global__ __launch_bounds__(256) void ncc_pass_w(
    const float* __restrict__ I, const float* __restrict__ J,
    float* __restrict__ P0, float* __restrict__ P1, float* __restrict__ P2,
    float* __restrict__ P3, float* __restrict__ P4) {
  const int lane = threadIdx.x & 31;
  const int wave = threadIdx.x >> 5;
  const int tile = blockIdx.x * 8 + wave;
  const int rowTile = tile / WTILES;
  const int wTile = tile - rowTile * WTILES;
  const int row0 = rowTile << 4;
  const int w0 = wTile << 4;

  const int mn = lane & 15;              // A row (M) / B col (N) for this lane
  const int khalf = (lane >> 4) << 1;    // lanes 0-15: K=0,1 ; 16-31: K=2,3
  const unsigned rowoff = (unsigned)(row0 + mn) * NCC_W;  // 32-bit elem offset
  const int wout = w0 + mn;

  v8f cI = {}, cJ = {}, cIJ = {}, cII = {}, cJJ = {};

#define NCC_W_CHUNK(BB0, BB1, OFF)                              \
  do {                                                          \
    const v2f iv = *(const v2f*)(I + (OFF));                    \
    const v2f jv = *(const v2f*)(J + (OFF));                    \
    const v2f bb = { (BB0), (BB1) };                            \
    cI  = wmma_f32_16x16x4(iv, bb, cI);                         \
    cJ  = wmma_f32_16x16x4(jv, bb, cJ);                         \
    cIJ = wmma_f32_16x16x4(iv * jv, bb, cIJ);                   \
    cII = wmma_f32_16x16x4(iv * iv, bb, cII);                   \
    cJJ = wmma_f32_16x16x4(jv * jv, bb, cJJ);                   \
  } while (0)

  {  // leading halo chunk: k = w0-4 .. w0-1, valid iff w0 > 0
    const bool v = (w0 > 0);
    const unsigned off = rowoff + (unsigned)(v ? (w0 - 4 + khalf) : 0);
    const int d0 = khalf - 4 - mn;
    NCC_W_CHUNK(v ? band9(d0) : 0.f, v ? band9(d0 + 1) : 0.f, off);
  }
  const unsigned mid = rowoff + (unsigned)(w0 + khalf);
#pragma unroll
  for (int c = 0; c < 4; ++c) {  // interior chunks: k = w0 .. w0+15
    const int d0 = 4 * c + khalf - mn;
    NCC_W_CHUNK(band9(d0), band9(d0 + 1), mid + (unsigned)(4 * c));
  }
  {  // trailing halo chunk: k = w0+16 .. w0+19, valid iff w0+16 < W
    const bool v = (w0 + 16) < NCC_W;
    const unsigned off = rowoff + (unsigned)(v ? (w0 + 16 + khalf) : (NCC_W - 2));
    const int d0 = 16 + khalf - mn;
    NCC_W_CHUNK(v ? band9(d0) : 0.f, v ? band9(d0 + 1) : 0.f, off);
  }
#undef NCC_W_CHUNK

  // D layout: VGPR r -> row r (lanes 0-15) / row r+8 (lanes 16-31), N = lane%16
  const int mb = (lane >> 4) << 3;
  const unsigned sidx = (unsigned)(row0 + mb) * NCC_W + (unsigned)wout;
#pragma unroll
  for (int r = 0; r < 8; ++r) {
    const unsigned idx = sidx + (unsigned)(r * NCC_W);
    P0[idx] = cI[r];
    P1[idx] = cJ[r];
    P2[idx] = cIJ[r];
    P3[idx] = cII[r];
    P4[idx] = cJJ[r];
  }
}

// ---------------------------------------------------------------------------
// Pass 2: 9-wide box sum along H for the 5 fields. M = 16 consecutive w
// (lane-coalesced A loads), N = 16 h outputs, K = h inputs [h0-4, h0+19].
// ---------------------------------------------------------------------------
__global__ __launch_bounds__(256) void ncc_pass_h(
    const float* __restrict__ P0, const float* __restrict__ P1,
    const float* __restrict__ P2, const float* __restrict__ P3,
    const float* __restrict__ P4,
    float* __restrict__ Q0, float* __restrict__ Q1, float* __restrict__ Q2,
    float* __restrict__ Q3, float* __restrict__ Q4) {
  const int lane = threadIdx.x & 31;
  const int wave = threadIdx.x >> 5;
  int t = blockIdx.x * 8 + wave;
  const int wTile = t % WTILES; t /= WTILES;
  const int hTile = t % HTILES; t /= HTILES;
  const int slice = t;  // b*D + d, 0..319
  const int w0 = wTile << 4;
  const int h0 = hTile << 4;
  const unsigned sbase = (unsigned)slice * NCC_HW;

  const int mn = lane & 15;
  const int khalf = (lane >> 4) << 1;
  const int hout = h0 + mn;

  v8f a0 = {}, a1 = {}, a2 = {}, a3 = {}, a4 = {};

#define NCC_H_CHUNK(BB0, BB1, O0)                                       \
  do {                                                                  \
    const unsigned o0 = (O0);                                           \
    const unsigned o1 = o0 + (unsigned)NCC_W;                           \
    const v2f bb = { (BB0), (BB1) };                                    \
    { v2f a = { P0[o0], P0[o1] }; a0 = wmma_f32_16x16x4(a, bb, a0); }   \
    { v2f a = { P1[o0], P1[o1] }; a1 = wmma_f32_16x16x4(a, bb, a1); }   \
    { v2f a = { P2[o0], P2[o1] }; a2 = wmma_f32_16x16x4(a, bb, a2); }   \
    { v2f a = { P3[o0], P3[o1] }; a3 = wmma_f32_16x16x4(a, bb, a3); }   \
    { v2f a = { P4[o0], P4[o1] }; a4 = wmma_f32_16x16x4(a, bb, a4); }   \
  } while (0)

  const unsigned cbase = sbase + (unsigned)(w0 + mn);
  {  // leading halo chunk: k = h0-4 .. h0-1, valid iff h0 > 0
    const bool v = (h0 > 0);
    const unsigned off = cbase + (unsigned)((v ? (h0 - 4 + khalf) : 0) * NCC_W);
    const int d0 = khalf - 4 - mn;
    NCC_H_CHUNK(v ? band9(d0) : 0.f, v ? band9(d0 + 1) : 0.f, off);
  }
  const unsigned midb = cbase + (unsigned)((h0 + khalf) * NCC_W);
#pragma unroll
  for (int c = 0; c < 4; ++c) {  // interior chunks: k = h0 .. h0+15
    const int d0 = 4 * c + khalf - mn;
    NCC_H_CHUNK(band9(d0), band9(d0 + 1), midb + (unsigned)(4 * c * NCC_W));
  }
  {  // trailing halo chunk: k = h0+16 .. h0+19, valid iff h0+16 < H
    const bool v = (h0 + 16) < NCC_H;
    const unsigned off = cbase + (unsigned)((v ? (h0 + 16 + khalf) : (NCC_H - 2)) * NCC_W);
    const int d0 = 16 + khalf - mn;
    NCC_H_CHUNK(v ? band9(d0) : 0.f, v ? band9(d0 + 1) : 0.f, off);
  }
#undef NCC_H_CHUNK

  // D tile: M = w (8 consecutive per lane, 32B aligned), N = h_out.
  const int mb = (lane >> 4) << 3;
  const unsigned idx = sbase + (unsigned)hout * NCC_W + (unsigned)(w0 + mb);
#define NCC_STORE_FIELD(ACC, QF)                                           \
  do {                                                                     \
    v4f lo = { (ACC)[0], (ACC)[1], (ACC)[2], (ACC)[3] };                   \
    v4f hi = { (ACC)[4], (ACC)[5], (ACC)[6], (ACC)[7] };                   \
    *(v4f*)((QF) + idx) = lo;                                              \
    *(v4f*)((QF) + idx + 4) = hi;                                          \
  } while (0)
  NCC_STORE_FIELD(a0, Q0);
  NCC_STORE_FIELD(a1, Q1);
  NCC_STORE_FIELD(a2, Q2);
  NCC_STORE_FIELD(a3, Q3);
  NCC_STORE_FIELD(a4, Q4);
#undef NCC_STORE_FIELD
}

// ---------------------------------------------------------------------------
// Pass 3: box sum along D via a 9-deep register shift window per column,
// fused NCC math + deterministic in-block tree reduction. All loads are
// saddr-form: shared 32-bit voffset per iteration + 5 SGPR field bases.
// ---------------------------------------------------------------------------
__global__ __launch_bounds__(256) void ncc_pass_d(
    const float* __restrict__ Q0, const float* __restrict__ Q1,
    const float* __restrict__ Q2, const float* __restrict__ Q3,
    const float* __restrict__ Q4, float* __restrict__ partials) {
  const int t = blockIdx.x * 256 + threadIdx.x;  // 0..COLS-1
  const int b = t / NCC_HW;
  const int hw = t - b * NCC_HW;
  const unsigned col = (unsigned)b * NCC_DHW + (unsigned)hw;

  float wI[9], wJ[9], wIJ[9], wII[9], wJJ[9];
#pragma unroll
  for (int i = 0; i < 9; ++i) { wI[i] = 0.f; wJ[i] = 0.f; wIJ[i] = 0.f; wII[i] = 0.f; wJJ[i] = 0.f; }
#pragma unroll
  for (int i = 0; i < 5; ++i) {  // prime window with d = 0..4
    const unsigned o = col + (unsigned)(i * NCC_HW);
    wI[4 + i] = Q0[o]; wJ[4 + i] = Q1[o]; wIJ[4 + i] = Q2[o];
    wII[4 + i] = Q3[o]; wJJ[4 + i] = Q4[o];
  }

  float local = 0.f;
  for (int d = 0; d < NCC_D; ++d) {
    float sI = 0.f, sJ = 0.f, sIJ = 0.f, sII = 0.f, sJJ = 0.f;
#pragma unroll
    for (int i = 0; i < 9; ++i) {
      sI += wI[i]; sJ += wJ[i]; sIJ += wIJ[i]; sII += wII[i]; sJJ += wJJ[i];
    }
    const float cross = sIJ - sI * sJ;
    const float iv = sII - sI * sI;
    const float jv = sJJ - sJ * sJ;
    local += cross / (sqrtf(iv * jv) + NCC_EPS);

    const int nd = d + 5;
#pragma unroll
    for (int i = 0; i < 8; ++i) {
      wI[i] = wI[i + 1]; wJ[i] = wJ[i + 1]; wIJ[i] = wIJ[i + 1];
      wII[i] = wII[i + 1]; wJJ[i] = wJJ[i + 1];
    }
    if (nd < NCC_D) {
      const unsigned o = col + (unsigned)nd * NCC_HW;
      wI[8] = Q0[o]; wJ[8] = Q1[o]; wIJ[8] = Q2[o]; wII[8] = Q3[o]; wJJ[8] = Q4[o];
    } else {
      wI[8] = 0.f; wJ[8] = 0.f; wIJ[8] = 0.f; wII[8] = 0.f; wJJ[8] = 0.f;
    }
  }

  __shared__ float red[256];
  red[threadIdx.x] = local;
  __syncthreads();
#pragma unroll
  for (int s = 128; s > 0; s >>= 1) {
    if (threadIdx.x < s) red[threadIdx.x] += red[threadIdx.x + s];
    __syncthreads();
  }
  if (threadIdx.x == 0) partials[blockIdx.x] = red[0];
}

// Fixed-order final reduction -> deterministic scalar output.
__global__ void ncc_finalize(const float* __restrict__ partials,
                             float* __restrict__ out, int n) {
  if (threadIdx.x == 0 && blockIdx.x == 0) {
    float s = 0.f;
    for (int i = 0; i < n; ++i) s += partials[i];
    out[0] = -s / (float)NCC_N;
  }
}

extern "C" void kernel_launch(void* const* d_in, const int* in_sizes, int n_in,
                              void* d_out, int out_size, void* d_ws, size_t ws_size,
                              hipStream_t stream) {
  (void)in_sizes; (void)n_in; (void)out_size; (void)ws_size;
  const float* I = (const float*)d_in[0];
  const float* J = (const float*)d_in[1];
  float* P = (float*)d_ws;                        // 5 * N floats (after W pass)
  float* Q = P + (size_t)5 * NCC_N;               // 5 * N floats (after H pass)
  float* partials = Q + (size_t)5 * NCC_N;        // COLS/256 = 240 floats
  const size_t FN = (size_t)NCC_N;

  ncc_pass_w<<<TILES_W / 8, 256, 0, stream>>>(
      I, J, P, P + FN, P + 2 * FN, P + 3 * FN, P + 4 * FN);
  ncc_pass_h<<<TILES_H / 8, 256, 0, stream>>>(
      P, P + FN, P + 2 * FN, P + 3 * FN, P + 4 * FN,
      Q, Q + FN, Q + 2 * FN, Q + 3 * FN, Q + 4 * FN);
  ncc_pass_d<<<COLS / 256, 256, 0, stream>>>(
      Q, Q + FN, Q + 2 * FN, Q + 3 * FN, Q + 4 * FN, partials);
  ncc_finalize<<<1, 32, 0, stream>>>(partials, (float*)d_out, COLS / 256);
}